// DynamicLstmGnn_30073361007040
// MI455X (gfx1250) — compile-verified
//
#include <hip/hip_runtime.h>

// KNN-graph for N x 64 fp32 points, k = 5, dense symmetrized N x N output.
// Phase A: row norms. Phase B: fused fp32-WMMA distance GEMM + per-row top-5.
// Phase C: zero 400MB output (bandwidth floor). Phase D: sparse symmetrize.

typedef float v2f __attribute__((ext_vector_type(2)));
typedef float v8f __attribute__((ext_vector_type(8)));

#define KD  64
#define KNN 5
#define NEG_INF (-3.402823466e38f)

// ---------------------------------------------------------------- row norms
__global__ __launch_bounds__(256) void row_norms_kernel(
    const float* __restrict__ x, float* __restrict__ xx, int n) {
  int i = blockIdx.x * 256 + threadIdx.x;
  if (i >= n) return;
  const float4* p = (const float4*)(x + (size_t)i * KD);
  float s = 0.f;
#pragma unroll
  for (int k = 0; k < KD / 4; ++k) {
    float4 v = p[k];
    s += v.x * v.x + v.y * v.y + v.z * v.z + v.w * v.w;
  }
  xx[i] = s;
}

// ------------------------------------------- fused WMMA distances + top-k
// One workgroup = 16 rows. 8 wave32s sweep 16-col tiles with stride 8.
// d(i,j) = 2*dot(xi,xj) - ||xi||^2 - ||xj||^2 accumulated via
// V_WMMA_F32_16X16X4_F32 (16 chained WMMAs cover K=64).
__global__ __launch_bounds__(256) void knn_topk_kernel(
    const float* __restrict__ x, const float* __restrict__ xx,
    float* __restrict__ topv, int* __restrict__ topi, int n) {
  // merge buffers: [row][wave][lane16][slot]; u16 indices keep LDS at 60 KB
  __shared__ float          sv[16][8][16][KNN];
  __shared__ unsigned short si[16][8][16][KNN];

  const int rowBase = blockIdx.x << 4;
  const int lane = threadIdx.x & 31;
  const int wave = threadIdx.x >> 5;
  const int half = lane >> 4;   // 0: K pair {0,1}, 1: K pair {2,3} per WMMA step
  const int l16  = lane & 15;

  // A tile (16 rows x 64): ISA 32-bit A layout — lane = M, vgpr pair = K
  v2f a[16];
  {
    const float* ap = x + (size_t)(rowBase + l16) * KD + 2 * half;
#pragma unroll
    for (int kk = 0; kk < 16; ++kk) a[kk] = *(const v2f*)(ap + 4 * kk);
  }
  // norms of this lane's 8 C-rows: M = g + 8*half
  float xr[8];
#pragma unroll
  for (int g = 0; g < 8; ++g) xr[g] = xx[rowBase + g + 8 * half];

  // private sorted (desc) top-5 per row; guarded insert keeps hot path cheap
  float tv[8][KNN];
  int   ti[8][KNN];
#pragma unroll
  for (int g = 0; g < 8; ++g)
#pragma unroll
    for (int s = 0; s < KNN; ++s) { tv[g][s] = NEG_INF; ti[g][s] = -1; }

  const int ntiles = n >> 4;
  for (int t = wave; t < ntiles; t += 8) {
    const int colBase = t << 4;
    const int col = colBase + l16;
    const float* bp = x + (size_t)col * KD + 2 * half;
    if (t + 8 < ntiles)  // next B tile for this wave -> global_prefetch path
      __builtin_prefetch(x + (size_t)(colBase + 128 + l16) * KD, 0, 1);

    v2f b[16];
#pragma unroll
    for (int kk = 0; kk < 16; ++kk) b[kk] = *(const v2f*)(bp + 4 * kk);

    v8f acc = {};
#pragma unroll
    for (int kk = 0; kk < 16; ++kk)
      acc = __builtin_amdgcn_wmma_f32_16x16x4_f32(
          false, a[kk], false, b[kk], (short)0, acc, false, false);

    const float xc = xx[col];
#pragma unroll
    for (int g = 0; g < 8; ++g) {
      float d = 2.0f * acc[g] - xr[g] - xc;
      if (d > tv[g][KNN - 1]) {          // rare path: insert + single bubble pass
        tv[g][KNN - 1] = d; ti[g][KNN - 1] = col;
#pragma unroll
        for (int s = KNN - 1; s > 0; --s) {
          if (tv[g][s] > tv[g][s - 1]) {
            float tf = tv[g][s - 1]; tv[g][s - 1] = tv[g][s]; tv[g][s] = tf;
            int   tt = ti[g][s - 1]; ti[g][s - 1] = ti[g][s]; ti[g][s] = tt;
          }
        }
      }
    }
  }

  // dump candidates (each column seen by exactly one lane -> no duplicates)
#pragma unroll
  for (int g = 0; g < 8; ++g) {
    int r = g + 8 * half;
#pragma unroll
    for (int s = 0; s < KNN; ++s) {
      sv[r][wave][l16][s] = tv[g][s];
      si[r][wave][l16][s] = (unsigned short)(ti[g][s] < 0 ? 0xFFFF : ti[g][s]);
    }
  }
  __syncthreads();

  // one thread per row merges its 640 candidates
  if (threadIdx.x < 16) {
    const int r = threadIdx.x;
    const float*          pv = &sv[r][0][0][0];
    const unsigned short* pi = &si[r][0][0][0];
    float bv[KNN]; int bi[KNN];
#pragma unroll
    for (int s = 0; s < KNN; ++s) { bv[s] = NEG_INF; bi[s] = -1; }
    for (int c = 0; c < 8 * 16 * KNN; ++c) {
      float d = pv[c];
      if (d > bv[KNN - 1]) {
        unsigned short ic = pi[c];
        bv[KNN - 1] = d; bi[KNN - 1] = (ic == 0xFFFF) ? -1 : (int)ic;
#pragma unroll
        for (int s = KNN - 1; s > 0; --s) {
          if (bv[s] > bv[s - 1]) {
            float tf = bv[s - 1]; bv[s - 1] = bv[s]; bv[s] = tf;
            int   tt = bi[s - 1]; bi[s - 1] = bi[s]; bi[s] = tt;
          }
        }
      }
    }
    const int row = rowBase + r;
#pragma unroll
    for (int s = 0; s < KNN; ++s) {
      topv[row * KNN + s] = bv[s];
      topi[row * KNN + s] = bi[s];
    }
  }
}

// ---------------------------------------------------------------- zero out
__global__ __launch_bounds__(256) void zero_kernel(float4* __restrict__ out,
                                                   long long n4) {
  long long i = (long long)blockIdx.x * 256 + threadIdx.x;
  const long long stride = (long long)gridDim.x * 256;
  float4 z; z.x = 0.f; z.y = 0.f; z.z = 0.f; z.w = 0.f;
  for (; i < n4; i += stride) out[i] = z;
}

// ------------------------------------------------------- sparse symmetrize
// w(i,j) = (vij + vji) / max(1, (vij>0)+(vji>0)); diagonal stays zero.
// Symmetric value -> racing duplicate writes from edge (j,i) are idempotent.
__global__ __launch_bounds__(256) void build_edges_kernel(
    const float* __restrict__ topv, const int* __restrict__ topi,
    float* __restrict__ out, int n) {
  int e = blockIdx.x * 256 + threadIdx.x;
  if (e >= n * KNN) return;
  const int i = e / KNN;
  const int j = topi[e];
  if (j < 0 || j == i) return;
  const float vij = topv[e];
  float vji = 0.0f;
#pragma unroll
  for (int t = 0; t < KNN; ++t)
    if (topi[j * KNN + t] == i) vji = topv[j * KNN + t];
  const int cnt = (vij > 0.f ? 1 : 0) + (vji > 0.f ? 1 : 0);
  const float denom = cnt ? (float)cnt : 1.0f;
  const float w = (vij + vji) / denom;
  out[(size_t)i * n + j] = w;
  out[(size_t)j * n + i] = w;
}

// ------------------------------------------------------------------ launch
extern "C" void kernel_launch(void* const* d_in, const int* in_sizes, int n_in,
                              void* d_out, int out_size, void* d_ws,
                              size_t ws_size, hipStream_t stream) {
  const float* x = (const float*)d_in[0];
  const int n = in_sizes[0] / KD;  // 10000 (multiple of 16)

  float* ws   = (float*)d_ws;
  float* xx   = ws;                                   // n floats
  float* topv = ws + (((size_t)n + 255) & ~(size_t)255);  // n*KNN floats
  int*   topi = (int*)(topv + (size_t)n * KNN);           // n*KNN ints
  float* out  = (float*)d_out;

  row_norms_kernel<<<(n + 255) / 256, 256, 0, stream>>>(x, xx, n);
  knn_topk_kernel<<<n / 16, 256, 0, stream>>>(x, xx, topv, topi, n);
  const long long n4 = ((long long)n * (long long)n) / 4;
  zero_kernel<<<8192, 256, 0, stream>>>((float4*)out, n4);
  build_edges_kernel<<<(n * KNN + 255) / 256, 256, 0, stream>>>(topv, topi,
                                                               out, n);
}